// LiteratiQuantRotated_61838939127936
// MI455X (gfx1250) — compile-verified
//
#include <hip/hip_runtime.h>
#include <hip/hip_bf16.h>
#include <stdint.h>

#define D_DIM   4096
#define GQ      128
#define NTOK    16384
#define NB4     1024           // 4-element blocks per row
#define NGRP    32             // quant groups per row
#define THREADS 256
#define NWAVES  8
#define ROWS_PB 8
#define EPS_F   1e-8f

typedef float v4f __attribute__((ext_vector_type(4)));

// Hamilton product q (x) v with component order (.x=w, .y=x, .z=y, .w=z)
__device__ __forceinline__ float4 qmul(const float4 q, const float4 v) {
    float4 r;
    r.x = q.x*v.x - q.y*v.y - q.z*v.z - q.w*v.w;
    r.y = q.x*v.y + q.y*v.x + q.z*v.w - q.w*v.z;
    r.z = q.x*v.z - q.y*v.w + q.z*v.x + q.w*v.y;
    r.w = q.x*v.w + q.y*v.z - q.z*v.y + q.w*v.x;
    return r;
}

__device__ __forceinline__ float4 sgn_scale(const float4 r, const float c) {
    float4 o;
    o.x = (r.x < 0.0f) ? -c : c;   // sign(0)/sign(-0) -> +1, matches reference
    o.y = (r.y < 0.0f) ? -c : c;
    o.z = (r.z < 0.0f) ? -c : c;
    o.w = (r.w < 0.0f) ? -c : c;
    return o;
}

__device__ __forceinline__ float4 scale4(const float4 v, const float s) {
    float4 o; o.x = v.x*s; o.y = v.y*s; o.z = v.z*s; o.w = v.w*s; return o;
}

// stream-once output: non-temporal b128 store (TH=NT) -> no L2 pollution.
// bit_cast keeps the existing register quad (no per-component v_mov shuffles).
__device__ __forceinline__ void nt_store4(const float4 v, float4* p) {
    __builtin_nontemporal_store(__builtin_bit_cast(v4f, v), (v4f*)p);
}

// CDNA5 async global->LDS 128-bit load (GV mode), tracked by ASYNCcnt
__device__ __forceinline__ void async_b128(uint32_t lds_off, uint64_t gaddr) {
    asm volatile("global_load_async_to_lds_b128 %0, %1, off"
                 :: "v"(lds_off), "v"(gaddr) : "memory");
}

extern "C" __global__ void __launch_bounds__(THREADS, 1)
isoquant_rot_kernel(const float* __restrict__ x, const float* __restrict__ q,
                    float* __restrict__ xhat, float* __restrict__ scales_out,
                    float* __restrict__ norms_out)
{
    __shared__ float4 xbuf[2][NB4];      // 2 x 16KB double-buffered row tiles
    __shared__ float  red[2][NWAVES];    // cross-wave sum-of-squares

    const int t    = threadIdx.x;
    const int lane = t & 31;
    const int w    = t >> 5;
    const int row0 = blockIdx.x * ROWS_PB;

    // quaternion blocks this thread owns (identical for every row) -> registers
    const float4* q4 = (const float4*)q;
    const float4  q0 = q4[t], q1 = q4[t + 256], q2 = q4[t + 512], q3 = q4[t + 768];
    const float4 cq0 = make_float4(q0.x, -q0.y, -q0.z, -q0.w);
    const float4 cq1 = make_float4(q1.x, -q1.y, -q1.z, -q1.w);
    const float4 cq2 = make_float4(q2.x, -q2.y, -q2.z, -q2.w);
    const float4 cq3 = make_float4(q3.x, -q3.y, -q3.z, -q3.w);

    const float4* x4 = (const float4*)x;

    // prologue: stage row0 into buffer 0
    {
        uint64_t g = (uint64_t)(uintptr_t)(x4 + (size_t)row0 * NB4);
#pragma unroll
        for (int k = 0; k < 4; ++k)
            async_b128((uint32_t)(uintptr_t)&xbuf[0][t + 256 * k],
                       g + (uint64_t)(t + 256 * k) * 16u);
    }

#pragma unroll 1
    for (int i = 0; i < ROWS_PB; ++i) {
        const int r   = row0 + i;
        const int cur = i & 1;

        // make sure previous LDS reads from the other buffer retired before
        // the async engine may overwrite it
        asm volatile("s_wait_dscnt 0" ::: "memory");

        if (i + 1 < ROWS_PB) {
            uint64_t g = (uint64_t)(uintptr_t)(x4 + (size_t)(r + 1) * NB4);
#pragma unroll
            for (int k = 0; k < 4; ++k)
                async_b128((uint32_t)(uintptr_t)&xbuf[cur ^ 1][t + 256 * k],
                           g + (uint64_t)(t + 256 * k) * 16u);
            // async loads retire in order: <=4 outstanding => current row ready
            asm volatile("s_wait_asynccnt 4" ::: "memory");
        } else {
            asm volatile("s_wait_asynccnt 0" ::: "memory");
        }

        // each wave reads exactly the LDS slice its own lanes wrote
        const float4 v0 = xbuf[cur][t];
        const float4 v1 = xbuf[cur][t + 256];
        const float4 v2 = xbuf[cur][t + 512];
        const float4 v3 = xbuf[cur][t + 768];

        float sq = v0.x*v0.x + v0.y*v0.y + v0.z*v0.z + v0.w*v0.w
                 + v1.x*v1.x + v1.y*v1.y + v1.z*v1.z + v1.w*v1.w
                 + v2.x*v2.x + v2.y*v2.y + v2.z*v2.z + v2.w*v2.w
                 + v3.x*v3.x + v3.y*v3.y + v3.z*v3.z + v3.w*v3.w;

        // forward rotation on raw x (linear: fold 1/norm into scales later)
        const float4 r0 = qmul(q0, v0);
        const float4 r1 = qmul(q1, v1);
        const float4 r2 = qmul(q2, v2);
        const float4 r3 = qmul(q3, v3);

        float a0 = fabsf(r0.x) + fabsf(r0.y) + fabsf(r0.z) + fabsf(r0.w);
        float a1 = fabsf(r1.x) + fabsf(r1.y) + fabsf(r1.z) + fabsf(r1.w);
        float a2 = fabsf(r2.x) + fabsf(r2.y) + fabsf(r2.z) + fabsf(r2.w);
        float a3 = fabsf(r3.x) + fabsf(r3.y) + fabsf(r3.z) + fabsf(r3.w);

        // wave32 butterfly: chunk k's lane-sum == quant group (w + 8k)
#pragma unroll
        for (int m = 16; m >= 1; m >>= 1) {
            sq += __shfl_xor(sq, m, 32);
            a0 += __shfl_xor(a0, m, 32);
            a1 += __shfl_xor(a1, m, 32);
            a2 += __shfl_xor(a2, m, 32);
            a3 += __shfl_xor(a3, m, 32);
        }

        if (lane == 0) red[cur][w] = sq;
        __syncthreads();
        float tot = red[cur][0] + red[cur][1] + red[cur][2] + red[cur][3]
                  + red[cur][4] + red[cur][5] + red[cur][6] + red[cur][7];
        const float nrm = fmaxf(__builtin_sqrtf(tot), EPS_F);

        // scales of the *normalized* rotated row: mean(|rot(x)|)/nrm
        const float s0 = (a0 * 0.0078125f) / nrm;
        const float s1 = (a1 * 0.0078125f) / nrm;
        const float s2 = (a2 * 0.0078125f) / nrm;
        const float s3 = (a3 * 0.0078125f) / nrm;

        if (lane == 0) {
            float* sp = scales_out + (size_t)r * NGRP + w;
            __builtin_nontemporal_store(s0, sp + 0);
            __builtin_nontemporal_store(s1, sp + 8);
            __builtin_nontemporal_store(s2, sp + 16);
            __builtin_nontemporal_store(s3, sp + 24);
        }
        if (t == 0) __builtin_nontemporal_store(nrm, norms_out + r);

        const float c0 = fmaxf(s0, EPS_F), c1 = fmaxf(s1, EPS_F);
        const float c2 = fmaxf(s2, EPS_F), c3 = fmaxf(s3, EPS_F);

        // dequant: conj-rotation of sign*scale, then restore norm
        const float4 o0 = scale4(qmul(cq0, sgn_scale(r0, c0)), nrm);
        const float4 o1 = scale4(qmul(cq1, sgn_scale(r1, c1)), nrm);
        const float4 o2 = scale4(qmul(cq2, sgn_scale(r2, c2)), nrm);
        const float4 o3 = scale4(qmul(cq3, sgn_scale(r3, c3)), nrm);

        float4* outp = (float4*)xhat + (size_t)r * NB4;
        nt_store4(o0, outp + t);
        nt_store4(o1, outp + t + 256);
        nt_store4(o2, outp + t + 512);
        nt_store4(o3, outp + t + 768);
    }
}

extern "C" void kernel_launch(void* const* d_in, const int* in_sizes, int n_in,
                              void* d_out, int out_size, void* d_ws, size_t ws_size,
                              hipStream_t stream) {
    (void)in_sizes; (void)n_in; (void)out_size; (void)d_ws; (void)ws_size;
    const float* x = (const float*)d_in[0];   // (16384, 4096) f32
    const float* q = (const float*)d_in[1];   // (1024, 4) f32 unit quaternions

    float* out        = (float*)d_out;
    float* xhat       = out;                                   // 16384*4096
    float* scales_out = xhat + (size_t)NTOK * D_DIM;           // 16384*32
    float* norms_out  = scales_out + (size_t)NTOK * NGRP;      // 16384

    dim3 grid(NTOK / ROWS_PB), block(THREADS);
    isoquant_rot_kernel<<<grid, block, 0, stream>>>(x, q, xhat, scales_out, norms_out);
}